// SoftmaxAttention_73521250173463
// MI455X (gfx1250) — compile-verified
//
#include <hip/hip_runtime.h>
#include <math.h>

// ---------------------------------------------------------------------------
// Problem constants (reference: codebook [4096,1,3,128,128] f32, anchor
// [1,3,128,128] f32, top_k=16, softmax temp 0.1, eps 1e-12).
// ---------------------------------------------------------------------------
#define NROWS    4096
#define DLEN     49152          // 3*128*128
#define TOPK     16
#define TEMP_INV 10.0f          // 1 / 0.1
#define EPSF     1e-12f

typedef float v2f __attribute__((ext_vector_type(2)));
typedef float v8f __attribute__((ext_vector_type(8)));

// ---------------------------------------------------------------------------
// Kernel 1: stream the whole codebook once. Each workgroup = 16 rows,
// 16 waves split the K dimension. Per wave-iteration:
//   A (16x4 f32 tile, ISA layout) <- one b64 load per lane from 16 rows
//   B (4x16)                      <- anchor chunk broadcast to all 16 columns
//   D += A x B  via V_WMMA_F32_16X16X4_F32  (column 0 = dot partials)
//   ssq accumulated with VALU FMAs on the same loaded values.
// Bandwidth-bound: 805 MB / 23.3 TB/s ~ 35 us.
// ---------------------------------------------------------------------------
__global__ __launch_bounds__(512) void sim_kernel(const float* __restrict__ cb,
                                                  const float* __restrict__ anchor,
                                                  float* __restrict__ sims) {
    __shared__ float s_dot[16][16];   // [wave][row]
    __shared__ float s_ssq[16][32];   // [wave][lane]

    const int tid  = threadIdx.x;
    const int lane = tid & 31;
    const int wave = tid >> 5;        // 0..15
    const int m    = lane & 15;       // row within tile
    const int hi   = lane >> 4;       // 0: K={0,1}, 1: K={2,3}
    const int row  = blockIdx.x * 16 + m;

    const float* __restrict__ arow = cb + (size_t)row * DLEN + (hi * 2);
    const float* __restrict__ bvec = anchor + (hi * 2);

    v8f c0 = {};
    v8f c1 = {};
    float ssq = 0.0f;

    // K chunks of 4 floats; wave w takes chunks c with (c & 15) == w.
    // Per-wave stride = 64 floats; unroll x2 => 128 floats per trip.
    int k = wave * 4;
    for (int it = 0; it < 384; ++it) {
        v2f a0 = *(const v2f*)(arow + k);
        v2f b0 = *(const v2f*)(bvec + k);
        v2f a1 = *(const v2f*)(arow + k + 64);
        v2f b1 = *(const v2f*)(bvec + k + 64);
        __builtin_prefetch(arow + k + 128, 0, 1);   // global_prefetch_b8

        ssq = fmaf(a0.x, a0.x, ssq);
        ssq = fmaf(a0.y, a0.y, ssq);
        ssq = fmaf(a1.x, a1.x, ssq);
        ssq = fmaf(a1.y, a1.y, ssq);

        // 8 args: (neg_a, A, neg_b, B, c_mod, C, reuse_a, reuse_b)
        c0 = __builtin_amdgcn_wmma_f32_16x16x4_f32(false, a0, false, b0,
                                                   (short)0, c0, false, false);
        c1 = __builtin_amdgcn_wmma_f32_16x16x4_f32(false, a1, false, b1,
                                                   (short)0, c1, false, false);
        k += 128;
    }
    v8f cs = c0 + c1;

    // Column 0 of D: lane 0 holds rows 0..7 in VGPRs 0..7, lane 16 rows 8..15.
    if (lane == 0) {
        #pragma unroll
        for (int i = 0; i < 8; ++i) s_dot[wave][i] = cs[i];
    } else if (lane == 16) {
        #pragma unroll
        for (int i = 0; i < 8; ++i) s_dot[wave][8 + i] = cs[i];
    }
    s_ssq[wave][lane] = ssq;
    __syncthreads();

    // Deterministic final combine (fixed summation order, no float atomics).
    if (tid < 16) {
        float dot = 0.0f, q = 0.0f;
        #pragma unroll
        for (int w = 0; w < 16; ++w) {
            dot += s_dot[w][tid];
            q   += s_ssq[w][tid] + s_ssq[w][tid + 16];
        }
        sims[blockIdx.x * 16 + tid] = dot / fmaxf(sqrtf(q), EPSF);
    }
}

// ---------------------------------------------------------------------------
// Kernel 2 (single block): anchor norm, scale sims, top-16 via 16
// deterministic argmax reduction rounds over an LDS-resident copy, softmax.
// ---------------------------------------------------------------------------
__global__ __launch_bounds__(256) void topk_kernel(const float* __restrict__ anchor,
                                                   const float* __restrict__ sims,
                                                   float* __restrict__ wts,
                                                   int* __restrict__ idxs) {
    __shared__ float vals[NROWS];     // 16 KB
    __shared__ float red[256];
    __shared__ int   redi[256];
    __shared__ float s_inv_na;
    __shared__ float topv[TOPK];
    __shared__ int   topi[TOPK];

    const int tid = threadIdx.x;

    // ||anchor||
    float q = 0.0f;
    for (int i = tid; i < DLEN; i += 256) {
        float x = anchor[i];
        q = fmaf(x, x, q);
    }
    red[tid] = q;
    __syncthreads();
    for (int s = 128; s > 0; s >>= 1) {
        if (tid < s) red[tid] += red[tid + s];
        __syncthreads();
    }
    if (tid == 0) s_inv_na = 1.0f / fmaxf(sqrtf(red[0]), EPSF);
    __syncthreads();

    const float inv_na = s_inv_na;
    for (int i = tid; i < NROWS; i += 256) vals[i] = sims[i] * inv_na;
    __syncthreads();

    for (int r = 0; r < TOPK; ++r) {
        float best = -INFINITY;
        int   bi   = NROWS;
        for (int i = tid; i < NROWS; i += 256) {
            float v = vals[i];
            if (v > best || (v == best && i < bi)) { best = v; bi = i; }
        }
        red[tid]  = best;
        redi[tid] = bi;
        __syncthreads();
        for (int s = 128; s > 0; s >>= 1) {
            if (tid < s) {
                float v2 = red[tid + s];
                int   i2 = redi[tid + s];
                if (v2 > red[tid] || (v2 == red[tid] && i2 < redi[tid])) {
                    red[tid] = v2; redi[tid] = i2;
                }
            }
            __syncthreads();
        }
        if (tid == 0) {
            topv[r] = red[0];
            topi[r] = redi[0];
            vals[redi[0]] = -INFINITY;
        }
        __syncthreads();
    }

    if (tid == 0) {
        const float mx = topv[0];           // descending -> max first
        float e[TOPK];
        float sum = 0.0f;
        #pragma unroll
        for (int kk = 0; kk < TOPK; ++kk) {
            e[kk] = __expf((topv[kk] - mx) * TEMP_INV);
            sum += e[kk];
        }
        const float inv = 1.0f / sum;
        #pragma unroll
        for (int kk = 0; kk < TOPK; ++kk) {
            wts[kk]  = e[kk] * inv;
            idxs[kk] = topi[kk];
        }
    }
}

// ---------------------------------------------------------------------------
// Kernel 3: out[d] = sum_k w[k] * codebook[idx[k]][d]   (3 MB gather, ~negligible)
// ---------------------------------------------------------------------------
__global__ __launch_bounds__(256) void blend_kernel(const float* __restrict__ cb,
                                                    const float* __restrict__ wts,
                                                    const int* __restrict__ idxs,
                                                    float* __restrict__ out) {
    __shared__ float  w[TOPK];
    __shared__ size_t base[TOPK];
    const int tid = threadIdx.x;
    if (tid < TOPK) {
        w[tid]    = wts[tid];
        base[tid] = (size_t)idxs[tid] * DLEN;
    }
    __syncthreads();

    const int d = blockIdx.x * 256 + tid;
    if (d < DLEN) {
        float acc = 0.0f;
        #pragma unroll
        for (int kk = 0; kk < TOPK; ++kk)
            acc = fmaf(w[kk], cb[base[kk] + d], acc);
        out[d] = acc;
    }
}

// ---------------------------------------------------------------------------
extern "C" void kernel_launch(void* const* d_in, const int* in_sizes, int n_in,
                              void* d_out, int out_size, void* d_ws, size_t ws_size,
                              hipStream_t stream) {
    const float* cb     = (const float*)d_in[0];  // [4096,1,3,128,128]
    const float* anchor = (const float*)d_in[1];  // [1,3,128,128]
    // d_in[2] = top_k (fixed at 16 by the reference setup)

    float* sims = (float*)d_ws;                   // 4096 floats
    float* wts  = sims + NROWS;                   // 16 floats
    int*   idxs = (int*)(wts + TOPK);             // 16 ints

    sim_kernel<<<NROWS / 16, 512, 0, stream>>>(cb, anchor, sims);
    topk_kernel<<<1, 256, 0, stream>>>(anchor, sims, wts, idxs);
    blend_kernel<<<(DLEN + 255) / 256, 256, 0, stream>>>(cb, wts, idxs, (float*)d_out);
}